// LegalizeDSPRAM_58737972740314
// MI455X (gfx1250) — compile-verified
//
#include <hip/hip_runtime.h>
#include <hip/hip_bf16.h>

// ---------------------------------------------------------------------------
// Greedy DSP/RAM legalization for gfx1250 (MI455X).
//  - Whole site table (S*8 bytes) + availability bitmask live in one WGP's
//    320KB LDS; the 4096-step serial scan never touches HBM in the hot loop.
//  - Sites staged global->LDS with CDNA5 async-to-LDS ops (ASYNCcnt path),
//    with a plain load + ds_store fallback if the builtin is absent.
//  - Masked argmin packed as u64 = (f32 dist bits << 32) | site index:
//    non-negative float bits order like floats, low index wins ties exactly
//    like jnp.argmin (first occurrence).
// ---------------------------------------------------------------------------

#if defined(__has_builtin)
#  if __has_builtin(__builtin_amdgcn_global_load_async_to_lds_b128)
#    define USE_ASYNC_LDS 1
#  endif
#endif
#ifndef USE_ASYNC_LDS
#  define USE_ASYNC_LDS 0
#endif

#if USE_ASYNC_LDS
// Builtin signature (from clang diagnostic): first param is
//   __attribute__((vector_size(16))) int __device__ *   (int4, addrspace(1))
// second param is the LDS-side int4 pointer (addrspace(3)).
typedef int async_v4i __attribute__((vector_size(16)));
typedef __attribute__((address_space(1))) async_v4i g_v4i;
typedef __attribute__((address_space(3))) async_v4i l_v4i;
#endif

__device__ __forceinline__ unsigned long long shfl_xor_u64(unsigned long long x, int m) {
    unsigned lo = (unsigned)(x & 0xFFFFFFFFull);
    unsigned hi = (unsigned)(x >> 32);
    lo = __shfl_xor(lo, m, 32);
    hi = __shfl_xor(hi, m, 32);
    return ((unsigned long long)hi << 32) | (unsigned long long)lo;
}

// ---- kernel 1: stable descending argsort of precondWL via rank counting ----
// rank(i) = #{ j : p[j] > p[i] } + #{ j : p[j] == p[i] && j < i }
// (matches jnp.argsort(-p) stable semantics); ranks are unique -> scatter.
__global__ void dsp_rank_kernel(const float* __restrict__ p,
                                int* __restrict__ order, int N) {
    __shared__ float sp[4096];
    const bool useLds = (N <= 4096);
    if (useLds) {
        for (int j = threadIdx.x; j < N; j += blockDim.x) sp[j] = p[j];
    }
    __syncthreads();
    int i = blockIdx.x * blockDim.x + threadIdx.x;
    if (i >= N) return;
    float pi = useLds ? sp[i] : p[i];
    int rank = 0;
    for (int j = 0; j < N; ++j) {
        float pj = useLds ? sp[j] : p[j];
        rank += (pj > pi) || (pj == pi && j < i);
    }
    order[rank] = i;
}

// ---- kernel 2: single-workgroup greedy scan, everything LDS-resident ------
__global__ __launch_bounds__(1024)
void dsp_greedy_kernel(const float* __restrict__ locX,
                       const float* __restrict__ locY,
                       const float* __restrict__ sites,   // [S,2] interleaved
                       const int*   __restrict__ order,
                       float* __restrict__ out,           // [2*N + 2]
                       int N, int S) {
    extern __shared__ __align__(16) char smem[];
    float* s_xy = (float*)smem;                                   // 2*S floats
    const int nw = (((S + 31) >> 5) + 1) & ~1;                    // even #words
    unsigned* s_avail = (unsigned*)(smem + (size_t)S * 8);        // nw words
    unsigned long long* s_wave =
        (unsigned long long*)(smem + (size_t)S * 8 + (size_t)nw * 4); // 32 u64

    const int tid  = threadIdx.x;
    const int lane = tid & 31;
    const int wid  = tid >> 5;
    const int NT   = 1024;

    // ---- stage site table into LDS ----
    const int totalFloats = S * 2;
#if USE_ASYNC_LDS
    {
        const char* gbase = (const char*)sites;
        char*       lbase = (char*)smem;
        const size_t totalBytes = (size_t)totalFloats * 4;
        const size_t vecBytes   = totalBytes & ~(size_t)15;
        for (size_t b = (size_t)tid * 16; b < vecBytes; b += (size_t)NT * 16) {
            __builtin_amdgcn_global_load_async_to_lds_b128(
                (g_v4i*)(gbase + b), (l_v4i*)(lbase + b), 0, 0);
        }
        for (int f = (int)(vecBytes / 4) + tid; f < totalFloats; f += NT)
            s_xy[f] = sites[f];
    }
#else
    for (int f = tid; f < totalFloats; f += NT) s_xy[f] = sites[f];
#endif

    // ---- availability bitmask: 1 = free ----
    for (int w = tid; w < nw; w += NT) {
        int base = w * 32;
        unsigned m = 0u;
        if (base + 32 <= S)      m = 0xFFFFFFFFu;
        else if (base < S)       m = (1u << (S - base)) - 1u;
        s_avail[w] = m;
    }

#if USE_ASYNC_LDS
#  if defined(__has_builtin) && __has_builtin(__builtin_amdgcn_s_wait_asynccnt)
    __builtin_amdgcn_s_wait_asynccnt(0);
#  else
    asm volatile("s_wait_asynccnt 0" ::: "memory");
#  endif
#endif
    __syncthreads();

    float maxMove = 0.0f;
    float sumMove = 0.0f;

    for (int k = 0; k < N; ++k) {
        const int inst = order[k];          // uniform -> scalar load
        const float lx = locX[inst];
        const float ly = locY[inst];

        // per-thread masked argmin over strided site chunk
        unsigned long long best = ~0ull;
        for (int j = tid; j < S; j += NT) {
            if ((s_avail[j >> 5] >> (j & 31)) & 1u) {
                float d = fabsf(lx - s_xy[2 * j]) + fabsf(ly - s_xy[2 * j + 1]);
                unsigned long long key =
                    ((unsigned long long)__float_as_uint(d) << 32) | (unsigned)j;
                best = key < best ? key : best;
            }
        }
        // wave32 xor-butterfly
        for (int m = 16; m > 0; m >>= 1) {
            unsigned long long o = shfl_xor_u64(best, m);
            best = o < best ? o : best;
        }
        if (lane == 0) s_wave[wid] = best;
        __syncthreads();

        if (wid == 0) {
            unsigned long long v = s_wave[lane];   // 32 wave partials
            for (int m = 16; m > 0; m >>= 1) {
                unsigned long long o = shfl_xor_u64(v, m);
                v = o < v ? o : v;
            }
            if (lane == 0) {
                const int j = (int)(unsigned)(v & 0xFFFFFFFFull);
                s_avail[j >> 5] &= ~(1u << (j & 31));   // retire site
                const float ox = s_xy[2 * j];
                const float oy = s_xy[2 * j + 1];
                out[inst]     = ox;                      // outX row
                out[N + inst] = oy;                      // outY row
                const float mv = fabsf(lx - ox) + fabsf(ly - oy);
                maxMove = fmaxf(maxMove, mv);
                sumMove += mv;
            }
        }
        __syncthreads();   // avail update visible before next step
    }

    if (tid == 0) {
        out[2 * N]     = maxMove;
        out[2 * N + 1] = sumMove / (float)N;
    }
}

extern "C" void kernel_launch(void* const* d_in, const int* in_sizes, int n_in,
                              void* d_out, int out_size, void* d_ws, size_t ws_size,
                              hipStream_t stream) {
    (void)n_in; (void)out_size; (void)ws_size;
    const float* locX  = (const float*)d_in[0];
    const float* locY  = (const float*)d_in[1];
    const float* prec  = (const float*)d_in[2];
    const float* sites = (const float*)d_in[3];
    const int N = in_sizes[0];
    const int S = in_sizes[3] / 2;

    int* order = (int*)d_ws;   // N ints of scratch

    dsp_rank_kernel<<<(N + 255) / 256, 256, 0, stream>>>(prec, order, N);

    const int nw = (((S + 31) >> 5) + 1) & ~1;
    const size_t smemBytes = (size_t)S * 8        // site table (x,y interleaved)
                           + (size_t)nw * 4       // availability bitmask
                           + 32 * 8;              // wave partials
    dsp_greedy_kernel<<<1, 1024, smemBytes, stream>>>(
        locX, locY, sites, order, (float*)d_out, N, S);
}